// Center_26860725469564
// MI455X (gfx1250) — compile-verified
//
#include <hip/hip_runtime.h>

// ---------------- tunables ----------------
#define WAVES          8
#define BLOCK          (WAVES * 32)          // 256 threads
#define DSLICE         16                    // floats of dim per block
#define CSTRIDE        17                    // padded LDS stride (bank-conflict break)
#define CMAX           1024                  // >= classes (1000)
#define NBUF           4                     // staging buffers
#define NAHEAD         3                     // buffers kept in flight
#define ROWS_PER_STEP  (WAVES * 16)          // 128 rows consumed per block step
#define INSTR_PER_BUF  2                     // async b128 per wave per buffer

// ============================================================================
// Kernel 1: per-(dim-slice, row-chunk) partial segment sums.
//   - rows staged global->LDS with async-to-LDS b128 copies (ASYNCcnt),
//     4-deep per-wave pipeline, no block barriers in the hot loop
//   - scatter-reduce with native LDS float atomics into a padded accumulator
//   - flush as plain stores to per-chunk partials (zero global atomics)
// ============================================================================
__global__ __launch_bounds__(BLOCK) void center_partial_kernel(
    const float* __restrict__ inputs,   // [n, dim]
    const int*   __restrict__ targets,  // [n]
    float* __restrict__ part_sums,      // [R][classes*dim]
    float* __restrict__ part_cnts,      // [R][classes]
    int n, int dim, int classes, int rows_per_chunk)
{
    __shared__ float acc[CMAX * CSTRIDE];                              // 69.6 KB
    __shared__ float cnt[CMAX];                                        //  4.0 KB
    __shared__ __align__(16) float stage[NBUF][WAVES][2][8 * DSLICE];  // 32.0 KB

    const int tid  = threadIdx.x;
    const int wave = tid >> 5;
    const int lane = tid & 31;
    const int ds   = blockIdx.x;             // 16-float dim slice
    const int rc   = blockIdx.y;             // row chunk
    const int d0   = ds * DSLICE;
    const int rowbase = rc * rows_per_chunk;

    // zero accumulators
    for (int i = tid; i < classes * CSTRIDE; i += BLOCK) acc[i] = 0.0f;
    if (ds == 0)
        for (int i = tid; i < classes; i += BLOCK) cnt[i] = 0.0f;
    __syncthreads();

    // lane -> (row-in-group, dim quad)
    const int rloc = lane >> 2;              // 0..7
    const int quad = lane & 3;               // 0..3 (4 floats)

    const int iters = rows_per_chunk / ROWS_PER_STEP;

    // stage 16 rows x 16 floats per wave per buffer (2 async b128 per wave)
    auto issue = [&](int it, int buf) {
        const int base = rowbase + it * ROWS_PER_STEP + wave * 16;
        const int goffA = ((base + rloc)     * dim + d0 + quad * 4) * 4;
        const int goffB = ((base + 8 + rloc) * dim + d0 + quad * 4) * 4;
        const unsigned laddrA =
            (unsigned)(size_t)&stage[buf][wave][0][rloc * DSLICE + quad * 4];
        const unsigned laddrB =
            (unsigned)(size_t)&stage[buf][wave][1][rloc * DSLICE + quad * 4];
        asm volatile("global_load_async_to_lds_b128 %0, %1, %2"
                     :: "v"(laddrA), "v"(goffA), "s"(inputs) : "memory");
        asm volatile("global_load_async_to_lds_b128 %0, %1, %2"
                     :: "v"(laddrB), "v"(goffB), "s"(inputs) : "memory");
    };

    // prologue: fill the pipeline
    for (int p = 0; p < NAHEAD && p < iters; ++p) issue(p, p);

    // preload class ids for step 0
    int cA = 0, cB = 0;
    if (iters > 0) {
        const int base = rowbase + wave * 16;
        cA = targets[base + rloc];
        cB = targets[base + 8 + rloc];
    }

    for (int it = 0; it < iters; ++it) {
        const int buf = it % NBUF;

        if (it + NAHEAD < iters)
            issue(it + NAHEAD, (it + NAHEAD) % NBUF);

        // preload next step's class ids BEFORE the async wait so the global
        // load latency overlaps it
        int nA = cA, nB = cB;
        if (it + 1 < iters) {
            const int base = rowbase + (it + 1) * ROWS_PER_STEP + wave * 16;
            nA = targets[base + rloc];
            nB = targets[base + 8 + rloc];
        }

        if (it + NAHEAD < iters) {
            // in-order completion: <= NAHEAD*INSTR_PER_BUF outstanding
            // guarantees the buffer we consume is done
            asm volatile("s_wait_asynccnt 6" ::: "memory");
        } else {
            asm volatile("s_wait_asynccnt 0" ::: "memory");
        }

        const float4 vA =
            *(const float4*)&stage[buf][wave][0][rloc * DSLICE + quad * 4];
        const float4 vB =
            *(const float4*)&stage[buf][wave][1][rloc * DSLICE + quad * 4];

        float* apA = &acc[cA * CSTRIDE + quad * 4];
        atomicAdd(&apA[0], vA.x);            // ds_add_f32
        atomicAdd(&apA[1], vA.y);
        atomicAdd(&apA[2], vA.z);
        atomicAdd(&apA[3], vA.w);

        float* apB = &acc[cB * CSTRIDE + quad * 4];
        atomicAdd(&apB[0], vB.x);
        atomicAdd(&apB[1], vB.y);
        atomicAdd(&apB[2], vB.z);
        atomicAdd(&apB[3], vB.w);

        if (ds == 0 && quad == 0) {
            atomicAdd(&cnt[cA], 1.0f);       // one count per row
            atomicAdd(&cnt[cB], 1.0f);
        }

        cA = nA; cB = nB;
    }

    __syncthreads();

    // flush this block's tile (plain coalesced stores)
    float* ps = part_sums + (size_t)rc * classes * dim;
    for (int i = tid; i < classes * DSLICE; i += BLOCK) {
        const int c = i >> 4;                // / DSLICE
        const int d = i & 15;                // % DSLICE
        ps[c * dim + d0 + d] = acc[c * CSTRIDE + d];
    }
    if (ds == 0) {
        float* pc = part_cnts + (size_t)rc * classes;
        for (int i = tid; i < classes; i += BLOCK) pc[i] = cnt[i];
    }
}

// ============================================================================
// Kernel 2: reduce R partials, divide by counts, fall back to class_weight.
// ============================================================================
__global__ __launch_bounds__(256) void center_finalize_kernel(
    const float* __restrict__ part_sums,    // [R][classes*dim]
    const float* __restrict__ part_cnts,    // [R][classes]
    const float* __restrict__ class_weight, // [classes*dim]
    float* __restrict__ out,                // [classes*dim]
    int classes, int dim, int R)
{
    const int dim4  = dim >> 2;
    const int total = classes * dim4;
    const int idx   = blockIdx.x * blockDim.x + threadIdx.x;
    if (idx >= total) return;

    const int c = idx / dim4;

    float cv = 0.0f;
    for (int r = 0; r < R; ++r) cv += part_cnts[(size_t)r * classes + c];

    float4 s = make_float4(0.f, 0.f, 0.f, 0.f);
    for (int r = 0; r < R; ++r) {
        const float4 p =
            ((const float4*)(part_sums + (size_t)r * classes * dim))[idx];
        s.x += p.x; s.y += p.y; s.z += p.z; s.w += p.w;
    }

    float4 o;
    if (cv > 0.0f) {
        const float inv = 1.0f / cv;
        o = make_float4(s.x * inv, s.y * inv, s.z * inv, s.w * inv);
    } else {
        o = ((const float4*)class_weight)[idx];
    }
    ((float4*)out)[idx] = o;
}

// ============================================================================
extern "C" void kernel_launch(void* const* d_in, const int* in_sizes, int n_in,
                              void* d_out, int out_size, void* d_ws, size_t ws_size,
                              hipStream_t stream) {
    const float* inputs       = (const float*)d_in[0];
    const int*   targets      = (const int*)d_in[1];
    // d_in[2] is the scalar `classes`; derived from shapes instead (host-side)
    const float* class_weight = (const float*)d_in[3];
    float*       out          = (float*)d_out;

    const int n       = in_sizes[1];              // 65536
    const int dim     = in_sizes[0] / n;          // 1024
    const int classes = in_sizes[3] / dim;        // 1000

    // choose row-chunk count R (power of two, <= 8) that fits in workspace
    const size_t per_chunk = (size_t)(classes * dim + classes) * sizeof(float);
    int R = 8;
    while (R > 1 && (size_t)R * per_chunk > ws_size) R >>= 1;
    const int rows_per_chunk = n / R;

    float* part_sums = (float*)d_ws;
    float* part_cnts = part_sums + (size_t)R * classes * dim;

    const int dslices = dim / DSLICE;             // 64
    dim3 grid1(dslices, R);
    center_partial_kernel<<<grid1, BLOCK, 0, stream>>>(
        inputs, targets, part_sums, part_cnts, n, dim, classes, rows_per_chunk);

    const int total4 = classes * (dim >> 2);
    center_finalize_kernel<<<(total4 + 255) / 256, 256, 0, stream>>>(
        part_sums, part_cnts, class_weight, out, classes, dim, R);
}